// LearnableAdpFilter_29291676958725
// MI455X (gfx1250) — compile-verified
//
#include <hip/hip_runtime.h>
#include <stdint.h>

// Problem shape (fixed by the reference: x = [16, 4, 512, 512] f32)
#define IMG_W 512
#define IMG_H 512
#define N_PLANES (16 * 4)

// Tiling: 64x16 outputs per 256-thread block, halo of 1 -> 66x18 LDS tile
#define TW 64
#define THH 16
#define LW (TW + 2)          // 66
#define LH (THH + 2)         // 18
#define LDS_ELEMS (LW * LH)  // 1188 floats = 4752 B

// min into a, max into b  (one v_min_f32 + one v_max_f32)
__device__ __forceinline__ void cswap(float& a, float& b) {
    float lo = fminf(a, b);
    b = fmaxf(a, b);
    a = lo;
}

__global__ __launch_bounds__(256)
void adp_filter_kernel(const float* __restrict__ in,
                       const float* __restrict__ nvp,
                       const float* __restrict__ nbp,
                       float* __restrict__ out) {
    __shared__ float tile[LDS_ELEMS];

    const int tid   = threadIdx.x;
    const int plane = blockIdx.z;
    const int x0    = blockIdx.x * TW;
    const int y0    = blockIdx.y * THH;

    const float* __restrict__ src = in  + (size_t)plane * (IMG_H * IMG_W);
    float* __restrict__       dst = out + (size_t)plane * (IMG_H * IMG_W);

    // Boundary tiles need zero padding in the halo: pre-zero LDS there.
    const bool boundary = (x0 == 0) | (y0 == 0) |
                          (x0 + TW >= IMG_W) | (y0 + THH >= IMG_H);
    if (boundary) {
        for (int i = tid; i < LDS_ELEMS; i += 256) tile[i] = 0.0f;
        __syncthreads();  // ds stores complete before async engine writes LDS
    }

    // Stage the (66 x 18) halo tile: CDNA5 async global -> LDS data mover.
    // Each in-bounds element is moved without a VGPR round trip; tracked by
    // ASYNCcnt. Out-of-bounds lanes are masked off (LDS already zeroed).
    for (int i = tid; i < LDS_ELEMS; i += 256) {
        const int ly = i / LW;
        const int lx = i - ly * LW;
        const int gx = x0 + lx - 1;
        const int gy = y0 + ly - 1;
        if ((unsigned)gx < IMG_W && (unsigned)gy < IMG_H) {
            const float* gp = src + gy * IMG_W + gx;
            const unsigned ldsoff = (unsigned)(uintptr_t)(&tile[i]);
            asm volatile("global_load_async_to_lds_b32 %0, %1, off"
                         :: "v"(ldsoff), "v"(gp)
                         : "memory");
        }
    }
    // Wait for this wave's async transfers, then workgroup barrier so every
    // wave's tile data is visible in LDS.
    asm volatile("s_wait_asynccnt 0x0" ::: "memory");
    __syncthreads();

    const float nv = nvp[0];
    const float nb = nbp[0];

    // 256 threads = 64 columns x 4 row-quads; each thread does 4 consecutive
    // rows of one column so the six LDS rows it touches are reused 4x.
    const int tx = tid & (TW - 1);
    const int r0 = (tid >> 6) * 4;   // 0,4,8,12
    const int lx = tx + 1;           // center column in LDS coords

    // Cache 6 rows x 3 cols of the window from LDS.
    float w[6][3];
#pragma unroll
    for (int r = 0; r < 6; ++r) {
        const float* rp = &tile[(r0 + r) * LW + (lx - 1)];
        w[r][0] = rp[0];
        w[r][1] = rp[1];
        w[r][2] = rp[2];
    }

#pragma unroll
    for (int k = 0; k < 4; ++k) {
        float p0 = w[k][0],     p1 = w[k][1],     p2 = w[k][2];
        float p3 = w[k + 1][0], p4 = w[k + 1][1], p5 = w[k + 1][2];
        float p6 = w[k + 2][0], p7 = w[k + 2][1], p8 = w[k + 2][2];

        const float cx = p4;  // original center pixel

        // Unbiased variance: (sum(x^2) - sum(x)^2/9) / 8
        const float s  = ((p0 + p1) + (p2 + p3)) + ((p4 + p5) + (p6 + p7)) + p8;
        float ss = p0 * p0;
        ss = fmaf(p1, p1, ss); ss = fmaf(p2, p2, ss);
        ss = fmaf(p3, p3, ss); ss = fmaf(p4, p4, ss);
        ss = fmaf(p5, p5, ss); ss = fmaf(p6, p6, ss);
        ss = fmaf(p7, p7, ss); ss = fmaf(p8, p8, ss);
        const float var = (ss - s * s * (1.0f / 9.0f)) * (1.0f / 8.0f);

        // Paeth 19-compare-exchange median-of-9 network.
        cswap(p1, p2); cswap(p4, p5); cswap(p7, p8);
        cswap(p0, p1); cswap(p3, p4); cswap(p6, p7);
        cswap(p1, p2); cswap(p4, p5); cswap(p7, p8);
        cswap(p0, p3); cswap(p5, p8); cswap(p4, p7);
        cswap(p3, p6); cswap(p1, p4); cswap(p2, p5);
        cswap(p4, p7); cswap(p4, p2); cswap(p6, p4);
        cswap(p4, p2);
        const float med = p4;

        const float y = cx - nv / (var + 1e-10f) * (cx - med + nb);
        dst[(size_t)(y0 + r0 + k) * IMG_W + (x0 + tx)] = fmaxf(y, 0.0f);
    }
}

extern "C" void kernel_launch(void* const* d_in, const int* in_sizes, int n_in,
                              void* d_out, int out_size, void* d_ws, size_t ws_size,
                              hipStream_t stream) {
    (void)in_sizes; (void)n_in; (void)out_size; (void)d_ws; (void)ws_size;

    const float* x  = (const float*)d_in[0];
    const float* nv = (const float*)d_in[1];
    const float* nb = (const float*)d_in[2];
    float* out      = (float*)d_out;

    dim3 grid(IMG_W / TW, IMG_H / THH, N_PLANES);  // (8, 32, 64)
    adp_filter_kernel<<<grid, 256, 0, stream>>>(x, nv, nb, out);
}